// SAGPooling_55533927137969
// MI455X (gfx1250) — compile-verified
//
#include <hip/hip_runtime.h>
#include <math.h>

#define N_NODES   50000
#define N_EDGES   400000
#define DIM       128
#define HEADS     4
#define HD        512              // HEADS*DIM
#define KTOP      40000
#define NEG_SLOPE 0.2f
#define TOTAL_MSG (N_EDGES + N_NODES)   // edges + self loops = 450000
#define SCAN_B    1024
#define SCAN_NB   ((N_EDGES + SCAN_B - 1) / SCAN_B)   // 391

typedef float v2f __attribute__((ext_vector_type(2)));
typedef float v8f __attribute__((ext_vector_type(8)));

// ---------------------------------------------------------------- init
__global__ void k_init(float* __restrict__ outbuf, float* __restrict__ m,
                       float* __restrict__ denom) {
    size_t stride = (size_t)gridDim.x * blockDim.x;
    size_t i = (size_t)blockIdx.x * blockDim.x + threadIdx.x;
    size_t n512 = (size_t)N_NODES * HD;
    for (size_t j = i; j < n512; j += stride) outbuf[j] = 0.0f;
    size_t n4 = (size_t)N_NODES * HEADS;
    for (size_t j = i; j < n4; j += stride) {
        m[j] = -__builtin_inff();
        denom[j] = 0.0f;
    }
}

// ---------------------------------------------------------------- GEMM h = x @ W (fp32 WMMA)
// One wave computes one 16x16 tile of h. 3125 m-tiles x 32 n-tiles = 100000 tiles
// = 12500 blocks x 8 waves, exact cover -> EXEC is all-1s everywhere (WMMA req).
__global__ void k_gemm(const float* __restrict__ x, const float* __restrict__ W,
                       float* __restrict__ h) {
    int wave = threadIdx.x >> 5;
    int lane = threadIdx.x & 31;
    int tile = blockIdx.x * 8 + wave;          // 0..99999
    int mt = tile >> 5;                        // tile / 32
    int nt = tile & 31;                        // tile % 32
    int m0 = mt << 4, n0 = nt << 4;
    int lo = lane & 15, hi = lane >> 4;

    const float* xrow = x + (size_t)(m0 + lo) * DIM;   // A: 16x4, row M=lo, K pair per hi
    const float* wcol = W + n0 + lo;                   // B: 4x16, col N=lo

    v8f c = {};
#pragma unroll
    for (int k0 = 0; k0 < DIM; k0 += 4) {
        v2f a, b;
        // 32-bit A 16x4 layout: VGPR0 = K=2*hi, VGPR1 = K=2*hi+1 (ISA 7.12.2)
        a.x = xrow[k0 + 2 * hi + 0];
        a.y = xrow[k0 + 2 * hi + 1];
        // 32-bit B 4x16 layout (mirror of A): rows K=2*hi, 2*hi+1 striped across lanes
        b.x = wcol[(size_t)(k0 + 2 * hi + 0) * HD];
        b.y = wcol[(size_t)(k0 + 2 * hi + 1) * HD];
        c = __builtin_amdgcn_wmma_f32_16x16x4_f32(false, a, false, b,
                                                  (short)0, c, false, false);
    }
    // C/D 16x16 layout: VGPR r -> row (r + 8*hi), col lo
    float* hb = h + (size_t)m0 * HD + n0 + lo;
#pragma unroll
    for (int r = 0; r < 8; ++r)
        hb[(size_t)(r + 8 * hi) * HD] = c[r];
}

// ---------------------------------------------------------------- a_s / a_d (one wave per node)
__global__ void k_attn(const float* __restrict__ h, const float* __restrict__ att_src,
                       const float* __restrict__ att_dst,
                       float* __restrict__ a_s, float* __restrict__ a_d) {
    int wave = threadIdx.x >> 5, lane = threadIdx.x & 31;
    int n = blockIdx.x * 8 + wave;             // 6250 blocks * 8 waves = 50000
    const float* hn = h + (size_t)n * HD;
#pragma unroll
    for (int hh = 0; hh < HEADS; ++hh) {
        float ss = 0.0f, sd = 0.0f;
        for (int j = lane; j < DIM; j += 32) {
            float v = hn[hh * DIM + j];
            ss += v * att_src[hh * DIM + j];
            sd += v * att_dst[hh * DIM + j];
        }
#pragma unroll
        for (int off = 16; off; off >>= 1) {
            ss += __shfl_xor(ss, off, 32);
            sd += __shfl_xor(sd, off, 32);
        }
        if (lane == 0) { a_s[n * HEADS + hh] = ss; a_d[n * HEADS + hh] = sd; }
    }
}

// ---------------------------------------------------------------- edge helpers
__device__ __forceinline__ void msg_ends(const int* __restrict__ ei, int e,
                                         int& src, int& dst) {
    if (e < N_EDGES) { src = ei[e]; dst = ei[N_EDGES + e]; }
    else             { src = dst = e - N_EDGES; }           // self loops
}

__device__ __forceinline__ float lrelu(float v) {
    return v > 0.0f ? v : NEG_SLOPE * v;
}

// f32 atomic-max via int/uint ordering trick (init = -inf)
__device__ __forceinline__ void atomicMaxF(float* addr, float v) {
    if (v >= 0.0f) atomicMax((int*)addr, __float_as_int(v));
    else           atomicMin((unsigned int*)addr, __float_as_uint(v));
}

// ---------------------------------------------------------------- segment max
__global__ void k_emax(const int* __restrict__ ei, const float* __restrict__ a_s,
                       const float* __restrict__ a_d, float* __restrict__ m) {
    int e = blockIdx.x * 256 + threadIdx.x;
    if (e >= TOTAL_MSG) return;
    int src, dst; msg_ends(ei, e, src, dst);
#pragma unroll
    for (int hh = 0; hh < HEADS; ++hh) {
        float v = lrelu(a_s[src * HEADS + hh] + a_d[dst * HEADS + hh]);
        atomicMaxF(&m[dst * HEADS + hh], v);
    }
}

// ---------------------------------------------------------------- segment exp-sum
__global__ void k_esum(const int* __restrict__ ei, const float* __restrict__ a_s,
                       const float* __restrict__ a_d, const float* __restrict__ m,
                       float* __restrict__ denom) {
    int e = blockIdx.x * 256 + threadIdx.x;
    if (e >= TOTAL_MSG) return;
    int src, dst; msg_ends(ei, e, src, dst);
#pragma unroll
    for (int hh = 0; hh < HEADS; ++hh) {
        float v = lrelu(a_s[src * HEADS + hh] + a_d[dst * HEADS + hh]);
        atomicAdd(&denom[dst * HEADS + hh], expf(v - m[dst * HEADS + hh]));
    }
}

// ---------------------------------------------------------------- weighted aggregation
// One 128-thread block per message; accumulator (102MB) is L2-resident so the
// f32 atomics resolve in L2, not HBM.
__global__ void k_agg(const int* __restrict__ ei, const float* __restrict__ a_s,
                      const float* __restrict__ a_d, const float* __restrict__ m,
                      const float* __restrict__ denom, const float* __restrict__ h,
                      float* __restrict__ outbuf) {
    int e = blockIdx.x;
    int t = threadIdx.x;                       // 0..127
    int src, dst; msg_ends(ei, e, src, dst);
    float alpha[HEADS];
#pragma unroll
    for (int hh = 0; hh < HEADS; ++hh) {
        float v = lrelu(a_s[src * HEADS + hh] + a_d[dst * HEADS + hh]);
        alpha[hh] = expf(v - m[dst * HEADS + hh]) / denom[dst * HEADS + hh];
    }
    const float* hs = h + (size_t)src * HD;
    float* od = outbuf + (size_t)dst * HD;
#pragma unroll
    for (int hh = 0; hh < HEADS; ++hh)
        atomicAdd(&od[hh * DIM + t], hs[hh * DIM + t] * alpha[hh]);
}

// ---------------------------------------------------------------- score = tanh(elu(gat) . w + b)
__global__ void k_score(const float* __restrict__ outbuf, const float* __restrict__ bias,
                        const float* __restrict__ lin_w, const float* __restrict__ lin_b,
                        float* __restrict__ score) {
    int wave = threadIdx.x >> 5, lane = threadIdx.x & 31;
    int n = blockIdx.x * 8 + wave;
    const float* on = outbuf + (size_t)n * HD;
    float s = 0.0f;
    for (int j = lane; j < HD; j += 32) {
        float v = on[j] + bias[j];
        v = (v > 0.0f) ? v : (expf(v) - 1.0f);     // elu(alpha=1)
        s += v * lin_w[j];
    }
#pragma unroll
    for (int off = 16; off; off >>= 1) s += __shfl_xor(s, off, 32);
    if (lane == 0) score[n] = tanhf(s + lin_b[0]);
}

// ---------------------------------------------------------------- exact stable rank (descending)
__global__ void k_rank(const float* __restrict__ score, int* __restrict__ node_map,
                       int* __restrict__ perm_i, float* __restrict__ perm_f) {
    __shared__ float sc[256];
    int i = blockIdx.x * 256 + threadIdx.x;
    float si = (i < N_NODES) ? score[i] : 0.0f;
    int cnt = 0;
    for (int base = 0; base < N_NODES; base += 256) {
        int j = base + threadIdx.x;
        sc[threadIdx.x] = (j < N_NODES) ? score[j] : -__builtin_inff();
        __syncthreads();
        int lim = N_NODES - base; if (lim > 256) lim = 256;
        for (int t = 0; t < lim; ++t) {
            float sj = sc[t];
            int j2 = base + t;
            cnt += (sj > si) || (sj == si && j2 < i);   // stable descending argsort rank
        }
        __syncthreads();
    }
    if (i < N_NODES) {
        int nm = (cnt < KTOP) ? cnt : -1;
        node_map[i] = nm;
        if (nm >= 0) { perm_i[nm] = i; perm_f[nm] = (float)i; }
    }
}

// ---------------------------------------------------------------- xr = x[perm] * score[perm]
__global__ void k_gather(const float* __restrict__ x, const float* __restrict__ score,
                         const int* __restrict__ perm_i, float* __restrict__ xr) {
    int k = blockIdx.x, t = threadIdx.x;       // blockDim = 128
    int p = perm_i[k];
    xr[(size_t)k * DIM + t] = x[(size_t)p * DIM + t] * score[p];
}

// ---------------------------------------------------------------- edge remap
__global__ void k_eflag(const int* __restrict__ ei, const int* __restrict__ node_map,
                        int* __restrict__ erow, int* __restrict__ ecol) {
    int e = blockIdx.x * 256 + threadIdx.x;
    if (e >= N_EDGES) return;
    erow[e] = node_map[ei[e]];
    ecol[e] = node_map[ei[N_EDGES + e]];
}

// exclusive scan of valid flags: per-block (1024 items) pass
__global__ void k_scanA(const int* __restrict__ erow, const int* __restrict__ ecol,
                        int* __restrict__ eprefix, int* __restrict__ bsum) {
    __shared__ int s[256];
    int b = blockIdx.x, t = threadIdx.x;
    int base = b * SCAN_B;
    int pre[4]; int loc = 0;
#pragma unroll
    for (int k = 0; k < 4; ++k) {
        int e = base + t * 4 + k;
        int val = 0;
        if (e < N_EDGES) val = (erow[e] >= 0 && ecol[e] >= 0);
        pre[k] = loc; loc += val;
    }
    s[t] = loc;
    __syncthreads();
    for (int off = 1; off < 256; off <<= 1) {
        int add = (t >= off) ? s[t - off] : 0;
        __syncthreads();
        s[t] += add;
        __syncthreads();
    }
    int texcl = (t == 0) ? 0 : s[t - 1];
#pragma unroll
    for (int k = 0; k < 4; ++k) {
        int e = base + t * 4 + k;
        if (e < N_EDGES) eprefix[e] = texcl + pre[k];
    }
    if (t == 255) bsum[b] = s[255];
}

// scan of 391 block sums (single block)
__global__ void k_scanB(const int* __restrict__ bsum, int* __restrict__ boff,
                        int* __restrict__ totalv) {
    __shared__ int s[512];
    int t = threadIdx.x;
    s[t] = (t < SCAN_NB) ? bsum[t] : 0;
    __syncthreads();
    for (int off = 1; off < 512; off <<= 1) {
        int add = (t >= off) ? s[t - off] : 0;
        __syncthreads();
        s[t] += add;
        __syncthreads();
    }
    if (t < SCAN_NB) boff[t] = (t == 0) ? 0 : s[t - 1];
    if (t == 511) *totalv = s[511];
}

// stable partition write: valid edges first (orig order), then -1-padded invalid
__global__ void k_ewrite(const int* __restrict__ erow, const int* __restrict__ ecol,
                         const int* __restrict__ eprefix, const int* __restrict__ boff,
                         const int* __restrict__ totalv, float* __restrict__ out_ei) {
    int e = blockIdx.x * 256 + threadIdx.x;
    if (e >= N_EDGES) return;
    int r = erow[e], c = ecol[e];
    int valid = (r >= 0 && c >= 0);
    int p = eprefix[e] + boff[e / SCAN_B];          // # valid edges before e
    int pos = valid ? p : (*totalv + (e - p));      // stable: invalid keep order after
    out_ei[pos]            = valid ? (float)r : -1.0f;
    out_ei[N_EDGES + pos]  = valid ? (float)c : -1.0f;
}

// ================================================================ launch
extern "C" void kernel_launch(void* const* d_in, const int* in_sizes, int n_in,
                              void* d_out, int out_size, void* d_ws, size_t ws_size,
                              hipStream_t stream) {
    const float* x       = (const float*)d_in[0];
    const int*   ei      = (const int*)  d_in[1];
    const float* W       = (const float*)d_in[2];
    const float* att_src = (const float*)d_in[3];
    const float* att_dst = (const float*)d_in[4];
    const float* gbias   = (const float*)d_in[5];
    const float* lin_w   = (const float*)d_in[6];
    const float* lin_b   = (const float*)d_in[7];

    float* outp   = (float*)d_out;
    float* xr     = outp;                                    // [K*D]
    float* out_ei = outp + (size_t)KTOP * DIM;               // [2*E]
    float* perm_f = out_ei + 2 * (size_t)N_EDGES;            // [K]

    // workspace carve-out (~210 MB)
    char* p = (char*)d_ws;
    auto carve = [&](size_t bytes) -> char* {
        char* r = p; p += (bytes + 255) & ~(size_t)255; return r;
    };
    float* h      = (float*)carve((size_t)N_NODES * HD * 4);
    float* ob     = (float*)carve((size_t)N_NODES * HD * 4);
    float* a_s    = (float*)carve((size_t)N_NODES * HEADS * 4);
    float* a_d    = (float*)carve((size_t)N_NODES * HEADS * 4);
    float* m      = (float*)carve((size_t)N_NODES * HEADS * 4);
    float* denom  = (float*)carve((size_t)N_NODES * HEADS * 4);
    float* score  = (float*)carve((size_t)N_NODES * 4);
    int*   nmap   = (int*)  carve((size_t)N_NODES * 4);
    int*   perm_i = (int*)  carve((size_t)KTOP * 4);
    int*   erow   = (int*)  carve((size_t)N_EDGES * 4);
    int*   ecol   = (int*)  carve((size_t)N_EDGES * 4);
    int*   eprefix= (int*)  carve((size_t)N_EDGES * 4);
    int*   bsum   = (int*)  carve((size_t)SCAN_NB * 4);
    int*   boff   = (int*)  carve((size_t)SCAN_NB * 4);
    int*   totalv = (int*)  carve(4);

    k_init <<<4096, 256, 0, stream>>>(ob, m, denom);
    k_gemm <<<12500, 256, 0, stream>>>(x, W, h);
    k_attn <<<6250, 256, 0, stream>>>(h, att_src, att_dst, a_s, a_d);
    k_emax <<<(TOTAL_MSG + 255) / 256, 256, 0, stream>>>(ei, a_s, a_d, m);
    k_esum <<<(TOTAL_MSG + 255) / 256, 256, 0, stream>>>(ei, a_s, a_d, m, denom);
    k_agg  <<<TOTAL_MSG, 128, 0, stream>>>(ei, a_s, a_d, m, denom, h, ob);
    k_score<<<6250, 256, 0, stream>>>(ob, gbias, lin_w, lin_b, score);
    k_rank <<<(N_NODES + 255) / 256, 256, 0, stream>>>(score, nmap, perm_i, perm_f);
    k_gather<<<KTOP, 128, 0, stream>>>(x, score, perm_i, xr);
    k_eflag<<<(N_EDGES + 255) / 256, 256, 0, stream>>>(ei, nmap, erow, ecol);
    k_scanA<<<SCAN_NB, 256, 0, stream>>>(erow, ecol, eprefix, bsum);
    k_scanB<<<1, 512, 0, stream>>>(bsum, boff, totalv);
    k_ewrite<<<(N_EDGES + 255) / 256, 256, 0, stream>>>(erow, ecol, eprefix, boff,
                                                        totalv, out_ei);
}